// PositionsAttentionMLP_36009005810129
// MI455X (gfx1250) — compile-verified
//
#include <hip/hip_runtime.h>
#include <math.h>

// Problem sizes (fixed by the reference)
constexpr int BB = 32;
constexpr int NN = 1024;
constexpr int HH = 128;
constexpr float EPSF  = 1e-5f;
constexpr float LOG2E = 1.4426950408889634f;

typedef __attribute__((ext_vector_type(2))) float v2f;
typedef __attribute__((ext_vector_type(8))) float v8f;

// XOR-butterfly across lanes via ds_swizzle_b32 (group-of-32 mode:
// offset = {xor_mask[14:10], or_mask[9:5]=0, and_mask[4:0]=0x1f}).
// Masks 1/2/4/8 stay inside each 16-lane half = the WMMA D-matrix
// column group we reduce over.
template <int XM>
__device__ __forceinline__ float swz_xor(float x) {
  return __int_as_float(
      __builtin_amdgcn_ds_swizzle(__float_as_int(x), (XM << 10) | 0x1f));
}

// ---------------------------------------------------------------------------
// Kernel 1: fold every H=128 reduction into 10 scalars.
// ---------------------------------------------------------------------------
__global__ void prep_kernel(const float* __restrict__ lin1_w,
                            const float* __restrict__ lin1_b,
                            const float* __restrict__ wp,
                            const float* __restrict__ bp,
                            const float* __restrict__ wd,
                            const float* __restrict__ bd,
                            const float* __restrict__ wc,
                            float* __restrict__ scal) {
  __shared__ float red[HH];
  const int h = threadIdx.x;  // 128 threads
  const float w1 = wc[h];
  const float w2 = wc[HH + h];
  const float v  = lin1_w[h] * w1;
  const float vb = lin1_b[h] * w1;
  float vals[10];
  vals[0] = v  * wp[h * 3 + 0];
  vals[1] = v  * wp[h * 3 + 1];
  vals[2] = v  * wp[h * 3 + 2];
  vals[3] = v  * bp[h];
  vals[4] = vb * wp[h * 3 + 0];
  vals[5] = vb * wp[h * 3 + 1];
  vals[6] = vb * wp[h * 3 + 2];
  vals[7] = vb * bp[h];
  vals[8] = wd[h] * w2;
  vals[9] = bd[h] * w2;
#pragma unroll
  for (int q = 0; q < 10; ++q) {
    red[h] = vals[q];
    __syncthreads();
    for (int s = HH / 2; s > 0; s >>= 1) {
      if (h < s) red[h] += red[h + s];
      __syncthreads();
    }
    if (h == 0) scal[q] = red[0];
    __syncthreads();
  }
}

// ---------------------------------------------------------------------------
// Kernel 2: per-node precompute — norm, |p|^2, A, Bv, new_norm.
// ---------------------------------------------------------------------------
__global__ void node_kernel(const float* __restrict__ pos,
                            const float* __restrict__ p,
                            const float* __restrict__ lin1_w,
                            const float* __restrict__ lin1_b,
                            const float* __restrict__ lin2_w,
                            const float* __restrict__ lin2_b,
                            const float* __restrict__ scal,
                            float* __restrict__ normW,
                            float* __restrict__ sqW,
                            float* __restrict__ AW,
                            float* __restrict__ BvW,
                            float* __restrict__ newnW) {
  const int n = blockIdx.x * blockDim.x + threadIdx.x;
  if (n >= BB * NN) return;
  const float x = pos[n * 3 + 0], y = pos[n * 3 + 1], z = pos[n * 3 + 2];
  const float qx = p[n * 3 + 0], qy = p[n * 3 + 1], qz = p[n * 3 + 2];
  const float nrm = sqrtf(x * x + y * y + z * z);
  const float sq  = qx * qx + qy * qy + qz * qz;
  const float A  = fmaf(x, scal[0], fmaf(y, scal[1], fmaf(z, scal[2], scal[3])));
  const float Bv = fmaf(x, scal[4], fmaf(y, scal[5], fmaf(z, scal[6], scal[7])));
  float acc = 0.f;
#pragma unroll 8
  for (int h = 0; h < HH; ++h) {
    const float hid = fmaf(nrm, lin1_w[h], lin1_b[h]);
    acc = fmaf(fmaxf(hid, 0.f), lin2_w[h], acc);
  }
  normW[n] = nrm;
  sqW[n]   = sq;
  AW[n]    = A;
  BvW[n]   = Bv;
  newnW[n] = acc + lin2_b[0];
}

// ---------------------------------------------------------------------------
// Kernel 3: fused flash-attention.
// ---------------------------------------------------------------------------
__global__ __launch_bounds__(256) void attn_kernel(
    const float* __restrict__ pos, const float* __restrict__ p,
    const unsigned char* __restrict__ node_mask,
    const unsigned char* __restrict__ pharma_mask,
    const float* __restrict__ bc, const float* __restrict__ scal,
    const float* __restrict__ normW, const float* __restrict__ sqW,
    const float* __restrict__ AW, const float* __restrict__ BvW,
    const float* __restrict__ newnW, float* __restrict__ out) {
  // Packed j-side LDS: one b128 per tile gives B-operand + |p|^2, one b128
  // gives values + pharma, one b64 gives the affine score terms.
  __shared__ float4 s_q2[NN];  // (-2px, -2py, -2pz, |p|^2)
  __shared__ float4 s_v[NN];   // (posx, posy, posz, pharma)
  __shared__ float2 s_ab[NN];  // (A*log2e, (Bv + dB + bc)*log2e)

  const int b   = blockIdx.y;
  const int tid = threadIdx.x;
  const float* __restrict__ pb   = p   + (size_t)b * NN * 3;
  const float* __restrict__ posb = pos + (size_t)b * NN * 3;

  const float cDL   = scal[8] * LOG2E;            // cD * log2e
  const float cnstL = (scal[9] + bc[0]) * LOG2E;  // (dB + bc) * log2e

  for (int n = tid; n < NN; n += 256) {
    const float qx = pb[n * 3 + 0], qy = pb[n * 3 + 1], qz = pb[n * 3 + 2];
    s_q2[n] = make_float4(-2.f * qx, -2.f * qy, -2.f * qz, sqW[b * NN + n]);
    s_v[n]  = make_float4(posb[n * 3 + 0], posb[n * 3 + 1], posb[n * 3 + 2],
                          (float)pharma_mask[b * NN + n]);
    s_ab[n] = make_float2(AW[b * NN + n] * LOG2E,
                          fmaf(BvW[b * NN + n], LOG2E, cnstL));
  }
  __syncthreads();

  const int wave = tid >> 5, lane = tid & 31;
  const int hi = lane >> 4, lj = lane & 15;
  const int i0 = blockIdx.x * 128 + wave * 16;

  // A operand: row i = i0 + (lane&15); lanes 0-15 hold K=0,1 (px,py),
  // lanes 16-31 hold K=2,3 (pz, |p|^2).
  const float4 qa = s_q2[i0 + lj];
  v2f amat;
  amat.x = hi ? (-0.5f * qa.z) : (-0.5f * qa.x);
  amat.y = hi ? qa.w           : (-0.5f * qa.y);

  float normr[8], pmic[8], pmid[8];
#pragma unroll
  for (int r = 0; r < 8; ++r) {
    const int i = i0 + r + 8 * hi;
    normr[r] = normW[b * NN + i];
    pmic[r]  = (float)pharma_mask[b * NN + i];
    pmid[r]  = (lj == r + 8 * hi) ? 0.f : pmic[r];  // diagonal folded in
  }

  float m[8], l[8], c0[8], c1[8], c2[8];
#pragma unroll
  for (int r = 0; r < 8; ++r) {
    m[r] = -1e30f; l[r] = 0.f; c0[r] = 0.f; c1[r] = 0.f; c2[r] = 0.f;
  }

  // One j-tile step: WMMA Gram tile + per-lane online softmax update.
  auto tile_step = [&](int jt, const float(&pmu)[8]) {
    const int j = jt * 16 + lj;
    const float4 q2 = s_q2[j];
    v2f bmat;
    bmat.x = hi ? q2.z : q2.x;
    bmat.y = hi ? 1.0f : q2.y;
    v8f cacc;
#pragma unroll
    for (int r = 0; r < 8; ++r) cacc[r] = q2.w;  // + |p_j|^2 per column
    // d2[i,j] = |p_i|^2 + |p_j|^2 - 2 p_i.p_j
    v8f d2 = __builtin_amdgcn_wmma_f32_16x16x4_f32(
        false, amat, false, bmat, (short)0, cacc, false, false);

    const float2 ab = s_ab[j];
    const float4 vv = s_v[j];
    const float cpj = cDL * vv.w;  // cD*log2e * pharma_j
#pragma unroll
    for (int r = 0; r < 8; ++r) {
      const float dist = __builtin_amdgcn_sqrtf(fmaxf(d2[r], 1e-12f));
      const float s = fmaf(pmu[r] * cpj, dist, fmaf(normr[r], ab.x, ab.y));
      const float mn = fmaxf(m[r], s);
      const float sc = __builtin_amdgcn_exp2f(m[r] - mn);
      const float pr = __builtin_amdgcn_exp2f(s - mn);
      m[r]  = mn;
      l[r]  = fmaf(l[r],  sc, pr);
      c0[r] = fmaf(c0[r], sc, pr * vv.x);
      c1[r] = fmaf(c1[r], sc, pr * vv.y);
      c2[r] = fmaf(c2[r], sc, pr * vv.z);
    }
  };

  // Only tile dt can contain the i==j diagonal; all others skip the mask.
  const int dt = i0 >> 4;
#pragma unroll 2
  for (int jt = 0; jt < dt; ++jt) tile_step(jt, pmic);
  tile_step(dt, pmid);
#pragma unroll 2
  for (int jt = dt + 1; jt < NN / 16; ++jt) tile_step(jt, pmic);

  // Merge (m,l,ctx) across the 16 lanes of each column group.
#define MERGE_STEP(XM)                                                        \
  {                                                                           \
    const float m2 = swz_xor<XM>(m[r]);                                       \
    const float l2 = swz_xor<XM>(l[r]);                                       \
    const float a0 = swz_xor<XM>(c0[r]);                                      \
    const float a1 = swz_xor<XM>(c1[r]);                                      \
    const float a2 = swz_xor<XM>(c2[r]);                                      \
    const float mn = fmaxf(m[r], m2);                                         \
    const float sa = __builtin_amdgcn_exp2f(m[r] - mn);                       \
    const float sb = __builtin_amdgcn_exp2f(m2 - mn);                         \
    m[r]  = mn;                                                               \
    l[r]  = l[r]  * sa + l2 * sb;                                             \
    c0[r] = c0[r] * sa + a0 * sb;                                             \
    c1[r] = c1[r] * sa + a1 * sb;                                             \
    c2[r] = c2[r] * sa + a2 * sb;                                             \
  }
#pragma unroll
  for (int r = 0; r < 8; ++r) {
    MERGE_STEP(1)
    MERGE_STEP(2)
    MERGE_STEP(4)
    MERGE_STEP(8)
  }
#undef MERGE_STEP

  // Lanes 0 and 16 now hold full rows r and r+8 respectively.
  if (lj == 0) {
#pragma unroll
    for (int r = 0; r < 8; ++r) {
      const int i = i0 + r + 8 * hi;
      const float inv = __builtin_amdgcn_rcpf(l[r]);
      const float nrm = normW[b * NN + i];
      const float nn  = newnW[b * NN + i];
      const float nm  = (float)node_mask[b * NN + i];
      const float f   = nn * nm / (nrm + EPSF);
      const size_t o  = ((size_t)b * NN + i) * 3;
      out[o + 0] = posb[i * 3 + 0] * f * (c0[r] * inv);
      out[o + 1] = posb[i * 3 + 1] * f * (c1[r] * inv);
      out[o + 2] = posb[i * 3 + 2] * f * (c2[r] * inv);
    }
  }
}

// ---------------------------------------------------------------------------
// Kernel 4: subtract per-batch mean over the node axis (in place).
// ---------------------------------------------------------------------------
__global__ void mean_kernel(float* __restrict__ out) {
  __shared__ float rx[256], ry[256], rz[256];
  const int b = blockIdx.x, t = threadIdx.x;
  float sx = 0.f, sy = 0.f, sz = 0.f;
  for (int n = t; n < NN; n += 256) {
    const size_t o = ((size_t)b * NN + n) * 3;
    sx += out[o + 0]; sy += out[o + 1]; sz += out[o + 2];
  }
  rx[t] = sx; ry[t] = sy; rz[t] = sz;
  __syncthreads();
  for (int s = 128; s > 0; s >>= 1) {
    if (t < s) { rx[t] += rx[t + s]; ry[t] += ry[t + s]; rz[t] += rz[t + s]; }
    __syncthreads();
  }
  const float mx = rx[0] * (1.f / NN);
  const float my = ry[0] * (1.f / NN);
  const float mz = rz[0] * (1.f / NN);
  for (int n = t; n < NN; n += 256) {
    const size_t o = ((size_t)b * NN + n) * 3;
    out[o + 0] -= mx; out[o + 1] -= my; out[o + 2] -= mz;
  }
}

extern "C" void kernel_launch(void* const* d_in, const int* in_sizes, int n_in,
                              void* d_out, int out_size, void* d_ws,
                              size_t ws_size, hipStream_t stream) {
  const float* pos            = (const float*)d_in[0];
  const unsigned char* nmask  = (const unsigned char*)d_in[1];
  const float* p              = (const float*)d_in[2];
  const unsigned char* pmask  = (const unsigned char*)d_in[3];
  const float* lin1_w         = (const float*)d_in[4];
  const float* lin1_b         = (const float*)d_in[5];
  const float* lin2_w         = (const float*)d_in[6];
  const float* lin2_b         = (const float*)d_in[7];
  const float* wp             = (const float*)d_in[8];
  const float* bp             = (const float*)d_in[9];
  const float* wd             = (const float*)d_in[10];
  const float* bd             = (const float*)d_in[11];
  const float* wc             = (const float*)d_in[12];
  const float* bc             = (const float*)d_in[13];
  float* out = (float*)d_out;

  float* wsf   = (float*)d_ws;
  float* scal  = wsf;                 // 16 floats
  float* normW = wsf + 16;            // B*N
  float* sqW   = normW + BB * NN;     // B*N
  float* AW    = sqW + BB * NN;       // B*N
  float* BvW   = AW + BB * NN;        // B*N
  float* newnW = BvW + BB * NN;       // B*N

  prep_kernel<<<1, HH, 0, stream>>>(lin1_w, lin1_b, wp, bp, wd, bd, wc, scal);
  node_kernel<<<(BB * NN) / 256, 256, 0, stream>>>(
      pos, p, lin1_w, lin1_b, lin2_w, lin2_b, scal, normW, sqW, AW, BvW, newnW);
  dim3 grid(NN / 128, BB);
  attn_kernel<<<grid, 256, 0, stream>>>(pos, p, nmask, pmask, bc, scal, normW,
                                        sqW, AW, BvW, newnW, out);
  mean_kernel<<<BB, 256, 0, stream>>>(out);
}